// LigerLMHeadCE_52467320488212
// MI455X (gfx1250) — compile-verified
//
#include <hip/hip_runtime.h>
#include <math.h>

// ---------------------------------------------------------------------------
// Fused LM-head + cross-entropy for MI455X (gfx1250).
// Memory-bound (weight = 206MB @ 23.3TB/s ~= 9.5us floor) -> keep fp32 and use
// V_WMMA_F32_16X16X4_F32; K loop software-pipelined with double-buffered
// register slabs (compiles to clause-batched global_load_b64 bursts with
// fine-grained s_wait_loadcnt before each wmma).
// ---------------------------------------------------------------------------

typedef __attribute__((ext_vector_type(2))) float v2f;
typedef __attribute__((ext_vector_type(8))) float v8f;

#define HDIM 1024
#define V_CHUNK 512          // columns of vocab per block (grid.y)
#define ROWS_PER_BLOCK 128   // 8 waves * 16 rows
#define IGNORE_INDEX (-100)

// log-sum-exp merge of two (max, sumexp) partial states
__device__ __forceinline__ void lse_merge(float& m, float& s, float om, float os) {
    float nm = fmaxf(m, om);
    float a0 = (s  > 0.f) ? s  * __expf(m  - nm) : 0.f;
    float a1 = (os > 0.f) ? os * __expf(om - nm) : 0.f;
    m = nm;
    s = a0 + a1;
}

// Load one K-slab (32 K values = 8 wmma steps) into register arrays.
__device__ __forceinline__ void load_slab(const float* __restrict__ a,
                                          const float* __restrict__ b,
                                          int k, v2f A[8], v2f B[8]) {
#pragma unroll
    for (int i = 0; i < 8; ++i) {
        A[i] = *(const v2f*)(a + k + 4 * i);
        B[i] = *(const v2f*)(b + k + 4 * i);
    }
}

// 8 chained f32 WMMAs over one K-slab.
__device__ __forceinline__ v8f mma_slab(const v2f A[8], const v2f B[8], v8f c) {
#pragma unroll
    for (int i = 0; i < 8; ++i) {
        c = __builtin_amdgcn_wmma_f32_16x16x4_f32(false, A[i], false, B[i],
                                                  (short)0, c, false, false);
    }
    return c;
}

// Pass 1: per (row-block, vocab-chunk): WMMA logits + online softmax partials.
// Partials stored [chunk][row] so the merge pass reads coalesced.
__global__ __launch_bounds__(256)
void ce_pass1(const float* __restrict__ x, const float* __restrict__ w,
              const int* __restrict__ target,
              float* __restrict__ pm, float* __restrict__ ps,
              float* __restrict__ picked, int N, int V, int nchunk)
{
    const int lane = threadIdx.x & 31;
    const int wave = threadIdx.x >> 5;
    const int lh   = lane & 15;        // column within tile / row owner
    const int hi   = lane >> 4;        // K-subgroup / row-half select
    const int kb   = hi * 2;           // K offset within each group of 4 (f32 WMMA layout)

    const int rowBase      = blockIdx.x * ROWS_PER_BLOCK + wave * 16;
    const int chunk        = blockIdx.y;
    const int colChunkBase = chunk * V_CHUNK;

    // Targets for the 8 rows this lane's accumulators cover (rows r + 8*hi).
    int tgt[8];
#pragma unroll
    for (int r = 0; r < 8; ++r) tgt[r] = target[rowBase + r + 8 * hi];

    // Per-lane online-softmax state for 8 rows.
    float m[8], s[8];
#pragma unroll
    for (int r = 0; r < 8; ++r) { m[r] = -__builtin_inff(); s[r] = 0.f; }

    // A-matrix (x): 16x4 f32 tile: lanes 0-15 -> rows, K=kb..kb+1 in two VGPRs.
    const float* aptr = x + (size_t)(rowBase + lh) * HDIM + kb;

    for (int t = 0; t < V_CHUNK / 16; ++t) {
        const int colTile = colChunkBase + t * 16;
        const int col     = colTile + lh;
        const int colLd   = (col < V) ? col : (V - 1);   // clamp OOB load addresses
        const float* bptr = w + (size_t)colLd * HDIM + kb;

        v8f c = {};
        // Software-pipelined K loop: double-buffered 32-K slabs so the next
        // slab's 16 loads are outstanding while the current 8 WMMAs execute.
        v2f A0[8], B0[8], A1[8], B1[8];
        load_slab(aptr, bptr, 0, A0, B0);
        int k;
        for (k = 0; k + 64 < HDIM; k += 64) {
            load_slab(aptr, bptr, k + 32, A1, B1);
            c = mma_slab(A0, B0, c);                 // slab k
            load_slab(aptr, bptr, k + 64, A0, B0);
            c = mma_slab(A1, B1, c);                 // slab k+32
        }
        // k == HDIM-64 here: A0/B0 hold slab HDIM-64; fetch and finish last slab.
        load_slab(aptr, bptr, HDIM - 32, A1, B1);
        c = mma_slab(A0, B0, c);
        c = mma_slab(A1, B1, c);

        // C layout: VGPR r -> row (r + 8*hi), column = colTile + lh.
        if (col < V) {
#pragma unroll
            for (int r = 0; r < 8; ++r) {
                float v  = c[r];
                float nm = fmaxf(m[r], v);
                s[r] = s[r] * __expf(m[r] - nm) + __expf(v - nm);
                m[r] = nm;
                if (tgt[r] == col) picked[rowBase + r + 8 * hi] = v;
            }
        }
    }

    // Merge (m,s) across the 16 lanes holding each row (butterfly within 16-group).
#pragma unroll
    for (int r = 0; r < 8; ++r) {
        float mm = m[r], ss = s[r];
#pragma unroll
        for (int off = 8; off >= 1; off >>= 1) {
            float om = __shfl_xor(mm, off, 32);
            float os = __shfl_xor(ss, off, 32);
            lse_merge(mm, ss, om, os);
        }
        if (lh == 0) {
            const int row = rowBase + r + 8 * hi;
            pm[(size_t)chunk * N + row] = mm;   // [chunk][row] layout
            ps[(size_t)chunk * N + row] = ss;
        }
    }
}

// Pass 2a: one thread per row; merge chunk partials (coalesced across threads),
// emit per-row loss and valid flag. No atomics -> deterministic.
__global__ __launch_bounds__(256)
void ce_rowmerge(const float* __restrict__ pm, const float* __restrict__ ps,
                 const float* __restrict__ picked, const int* __restrict__ target,
                 float* __restrict__ loss, float* __restrict__ cnt,
                 int N, int nchunk)
{
    const int row = blockIdx.x * 256 + threadIdx.x;
    if (row >= N) return;
    float mm = -__builtin_inff(), ss = 0.f;
    for (int c = 0; c < nchunk; ++c) {
        lse_merge(mm, ss, pm[(size_t)c * N + row], ps[(size_t)c * N + row]);
    }
    const float logz  = mm + __logf(ss);
    const bool  valid = (target[row] != IGNORE_INDEX);
    loss[row] = valid ? (logz - picked[row]) : 0.f;
    cnt[row]  = valid ? 1.f : 0.f;
}

// Pass 2b: deterministic tree-reduction of per-row losses -> scalar mean.
__global__ __launch_bounds__(1024)
void ce_final(const float* __restrict__ loss, const float* __restrict__ cnt,
              float* __restrict__ out, int N)
{
    __shared__ float s_sum[1024];
    __shared__ float s_cnt[1024];
    float lsum = 0.f, lcnt = 0.f;
    for (int row = threadIdx.x; row < N; row += 1024) {
        lsum += loss[row];
        lcnt += cnt[row];
    }
    s_sum[threadIdx.x] = lsum;
    s_cnt[threadIdx.x] = lcnt;
    __syncthreads();
    for (int o = 512; o > 0; o >>= 1) {
        if ((int)threadIdx.x < o) {
            s_sum[threadIdx.x] += s_sum[threadIdx.x + o];
            s_cnt[threadIdx.x] += s_cnt[threadIdx.x + o];
        }
        __syncthreads();
    }
    if (threadIdx.x == 0) out[0] = s_sum[0] / fmaxf(s_cnt[0], 1.f);
}

extern "C" void kernel_launch(void* const* d_in, const int* in_sizes, int n_in,
                              void* d_out, int out_size, void* d_ws, size_t ws_size,
                              hipStream_t stream) {
    const float* x      = (const float*)d_in[0];   // [N, H]
    const int*   target = (const int*)d_in[1];     // [N]
    const float* w      = (const float*)d_in[2];   // [V, H]

    const int N = in_sizes[0] / HDIM;              // 4096
    const int V = in_sizes[2] / HDIM;              // 50257
    const int nchunk = (V + V_CHUNK - 1) / V_CHUNK;

    float* pm     = (float*)d_ws;                  // [nchunk, N] running max
    float* ps     = pm + (size_t)N * nchunk;       // [nchunk, N] running sumexp
    float* picked = ps + (size_t)N * nchunk;       // [N] target logit
    float* loss   = picked + N;                    // [N] per-row loss
    float* cnt    = loss + N;                      // [N] per-row valid flag

    dim3 grid(N / ROWS_PER_BLOCK, nchunk);         // x-major => row-blocks of a
                                                   // chunk co-resident, weight
                                                   // slice shared via L2
    ce_pass1<<<grid, 256, 0, stream>>>(x, w, target, pm, ps, picked, N, V, nchunk);
    ce_rowmerge<<<(N + 255) / 256, 256, 0, stream>>>(pm, ps, picked, target,
                                                     loss, cnt, N, nchunk);
    ce_final<<<1, 1024, 0, stream>>>(loss, cnt, (float*)d_out, N);
}